// HierarchicalPoolingLayer_48533130445229
// MI455X (gfx1250) — compile-verified
//
#include <hip/hip_runtime.h>
#include <hip/hip_bf16.h>
#include <math.h>

#define NODES 50000
#define EDGES 600000
#define DIM   128
#define DEA   16
#define KEEP  25000
#define LN_EPS 1e-5f

typedef __attribute__((ext_vector_type(2))) float v2f;
typedef __attribute__((ext_vector_type(8))) float v8f;
typedef __attribute__((ext_vector_type(4))) unsigned int u32x4;
typedef __attribute__((ext_vector_type(4))) int i32x4;
typedef __attribute__((ext_vector_type(8))) int i32x8;

// ---------------------------------------------------------------- init
__global__ __launch_bounds__(256) void k0_init(int* __restrict__ mapping,
                                               unsigned* __restrict__ hist,
                                               unsigned* __restrict__ meta) {
  const int i = blockIdx.x * 256 + threadIdx.x;
  if (i < NODES) mapping[i] = -1;
  if (i < 4096)  hist[i] = 0u;
  if (i < 8)     meta[i] = 0u;
}

// ---------------------------------------------------------------- scores: one wave per node
__global__ __launch_bounds__(256) void k1_score(const float* __restrict__ x,
                                                const float* __restrict__ wrel,
                                                const float* __restrict__ wroot,
                                                const float* __restrict__ brel,
                                                float* __restrict__ srel,
                                                float* __restrict__ score) {
  const int wv   = (blockIdx.x * 256 + threadIdx.x) >> 5;  // node id
  const int lane = threadIdx.x & 31;
  if (wv >= NODES) return;
  const float4 xv = *(const float4*)(x + (long long)wv * DIM + lane * 4);
  const float4 wr = *(const float4*)(wrel + lane * 4);
  const float4 wo = *(const float4*)(wroot + lane * 4);
  float r1 = xv.x * wr.x + xv.y * wr.y + xv.z * wr.z + xv.w * wr.w;
  float r2 = xv.x * wo.x + xv.y * wo.y + xv.z * wo.z + xv.w * wo.w;
#pragma unroll
  for (int off = 16; off > 0; off >>= 1) {
    r1 += __shfl_xor(r1, off, 32);
    r2 += __shfl_xor(r2, off, 32);
  }
  if (lane == 0) {
    srel[wv]  = r1;
    score[wv] = r2 + brel[0];
  }
}

// ---------------------------------------------------------------- edge scatter-add
__global__ __launch_bounds__(256) void k2_scatter(const int* __restrict__ ei,
                                                  const float* __restrict__ srel,
                                                  float* __restrict__ score) {
  const int e = blockIdx.x * 256 + threadIdx.x;
  if (e < EDGES) {
    const int s = ei[e];
    const int d = ei[EDGES + e];
    atomicAdd(score + d, srel[s]);   // global_atomic_add_f32
  }
}

// ---------------------------------------------------------------- monotone keys + histogram
__global__ __launch_bounds__(256) void k3_key(const float* __restrict__ score,
                                              unsigned* __restrict__ key,
                                              unsigned* __restrict__ hist) {
  const int n = blockIdx.x * 256 + threadIdx.x;
  if (n < NODES) {
    unsigned u = __float_as_uint(score[n]);
    u = (u >> 31) ? ~u : (u | 0x80000000u);   // ascending-monotone map
    key[n] = u;
    atomicAdd(hist + (u >> 20), 1u);
  }
}

// ---------------------------------------------------------------- radix-select threshold bin
__global__ __launch_bounds__(256) void k4_select(const unsigned* __restrict__ hist,
                                                 unsigned* __restrict__ meta) {
  __shared__ unsigned part[256];
  const int t = threadIdx.x;
  unsigned s = 0;
#pragma unroll
  for (int j = 0; j < 16; ++j) s += hist[t * 16 + j];
  part[t] = s;
  __syncthreads();
  if (t == 0) {
    unsigned cum = 0;
    int g = 255;
    for (; g >= 0; --g) {
      if (cum + part[g] >= (unsigned)KEEP) break;
      cum += part[g];
    }
    int bstar = 0;
    if (g >= 0) {
      for (int b = g * 16 + 15; b >= g * 16; --b) {
        cum += hist[b];
        if (cum >= (unsigned)KEEP) { bstar = b; break; }
      }
    }
    meta[0] = (unsigned)bstar;
  }
}

// ---------------------------------------------------------------- compact candidates
__global__ __launch_bounds__(256) void k5_compact(const unsigned* __restrict__ key,
                                                  unsigned* __restrict__ meta,
                                                  unsigned* __restrict__ ckey,
                                                  int* __restrict__ cidx) {
  const int n = blockIdx.x * 256 + threadIdx.x;
  if (n < NODES) {
    const unsigned k = key[n];
    if ((k >> 20) >= meta[0]) {
      const unsigned p = atomicAdd(meta + 2, 1u);
      ckey[p] = k;
      cidx[p] = n;
    }
  }
}

// ---------------------------------------------------------------- exact rank among candidates
__global__ __launch_bounds__(256) void k6_rank(const unsigned* __restrict__ ckey,
                                               const int* __restrict__ cidx,
                                               const unsigned* __restrict__ meta,
                                               const float* __restrict__ score,
                                               int* __restrict__ perm,
                                               float* __restrict__ tsc,
                                               int* __restrict__ mapping) {
  __shared__ unsigned sk[1024];
  __shared__ int      si[1024];
  const int t = threadIdx.x;
  const int i = blockIdx.x * 256 + t;
  const unsigned M = meta[2];
  const bool act = (unsigned)i < M;
  unsigned myk = 0;
  int myi = 0;
  if (act) { myk = ckey[i]; myi = cidx[i]; }
  unsigned rank = 0;
  for (unsigned base = 0; base < M; base += 1024u) {
    unsigned cnt = M - base;
    if (cnt > 1024u) cnt = 1024u;
    for (unsigned j = t; j < cnt; j += 256u) {
      sk[j] = ckey[base + j];
      si[j] = cidx[base + j];
    }
    __syncthreads();
    if (act) {
      for (unsigned j = 0; j < cnt; ++j) {
        const unsigned kj = sk[j];
        const int ij = si[j];
        // descending by score; tie -> lower node index first (matches lax.top_k)
        rank += (unsigned)((kj > myk) || (kj == myk && ij < myi));
      }
    }
    __syncthreads();
  }
  if (act && rank < (unsigned)KEEP) {
    perm[rank] = myi;
    tsc[rank]  = score[myi];
    mapping[myi] = (int)rank;
  }
}

// ---------------------------------------------------------------- gather + gate + GEMM (f32 WMMA) + ReLU + LayerNorm
__global__ __launch_bounds__(256) void k7_proj(const float* __restrict__ x,
                                               const float* __restrict__ Wp,
                                               const float* __restrict__ bproj,
                                               const float* __restrict__ gamma,
                                               const float* __restrict__ beta,
                                               const int* __restrict__ perm,
                                               const float* __restrict__ tsc,
                                               float* __restrict__ xout) {
  __shared__ float sW[128 * 132];  // W_proj row-major, TDM-padded stride 132 (bank-conflict-free)
  __shared__ float sA[16 * 132];   // gathered+gated x rows
  __shared__ float sH[16 * 132];   // relu(GEMM+bias) tile for LayerNorm
  const int t  = threadIdx.x;
  const int m0 = blockIdx.x * 16;

  // --- TDM: DMA W_proj (128x128 f32) into LDS with 4-DWORD pad every 128 DWORDs (stride 132)
  if (t < 32) {
    const unsigned long long ga = (unsigned long long)Wp;
    const unsigned ldsoff = (unsigned)(unsigned long long)(void*)&sW[0];
    u32x4 g0;
    g0.x = 1u;                                                 // count = 1 valid descriptor
    g0.y = ldsoff;                                             // lds_addr
    g0.z = (unsigned)(ga & 0xFFFFFFFFull);                     // global_addr[31:0]
    g0.w = (unsigned)((ga >> 32) & 0x1FFFFFFull) | (2u << 30); // global_addr[56:32] | type=2
    i32x8 g1;
    g1[0] = (int)((2u << 16)     // data_size = 4B
                | (1u << 20)     // pad_enable
                | (6u << 22)     // pad_interval: every 128 DWORDs
                | (3u << 25));   // pad_amount: 4 DWORDs -> row stride 132
    g1[1] = (int)(128u << 16);   // tensor_dim0 = 128
    g1[2] = (int)(128u << 16);   // tensor_dim1 = 128
    g1[3] = (int)(128u << 16);   // tile_dim0 = 128
    g1[4] = 128;                 // tile_dim1 = 128
    g1[5] = 128;                 // tensor_dim0_stride = 128
    g1[6] = (int)(128u << 16);   // tensor_dim1_stride low bits
    g1[7] = 0;
    i32x4 g2; g2[0] = 0; g2[1] = 0; g2[2] = 0; g2[3] = 0;
    i32x4 g3 = g2;
    i32x8 g4;                    // unused descriptor group (6-arg toolchain form)
    g4[0] = 0; g4[1] = 0; g4[2] = 0; g4[3] = 0;
    g4[4] = 0; g4[5] = 0; g4[6] = 0; g4[7] = 0;
    __builtin_amdgcn_tensor_load_to_lds(g0, g1, g2, g3, g4, 0);
  }

  // --- stage A tile (overlaps the TDM): 16 gathered rows, gated by tanh(score)
  {
    const int e0  = t * 8;
    const int row = e0 >> 7;
    const int col = e0 & 127;
    const int i   = m0 + row;
    float4 p0 = {0.f, 0.f, 0.f, 0.f}, p1 = p0;
    if (i < KEEP) {
      const int node   = perm[i];
      const float gate = tanhf(tsc[i]);
      const float4* xs = (const float4*)(x + (long long)node * DIM + col);
      p0 = xs[0]; p1 = xs[1];
      p0.x *= gate; p0.y *= gate; p0.z *= gate; p0.w *= gate;
      p1.x *= gate; p1.y *= gate; p1.z *= gate; p1.w *= gate;
    }
    float4* dst = (float4*)(sA + row * 132 + col);
    dst[0] = p0; dst[1] = p1;
  }
  if (t < 32) __builtin_amdgcn_s_wait_tensorcnt(0);
  __syncthreads();

  // --- one 16x16 f32 C tile per wave; 8 waves cover all 128 output cols
  const int w    = t >> 5;
  const int lane = t & 31;
  const int ml   = lane & 15;
  const int koff = (lane < 16) ? 0 : 2;   // ISA 16x4 f32 A/B layout: lane halves hold K pairs
  const int ocol = w * 16 + ml;

  const float* __restrict__ Ap = sA + ml * 132 + koff;
  const float* __restrict__ Bp = sW + ocol * 132 + koff;   // B[k][n] = W[n][k]

  v8f acc = {0.f, 0.f, 0.f, 0.f, 0.f, 0.f, 0.f, 0.f};
#pragma unroll
  for (int kk = 0; kk < 32; ++kk) {      // 32 x (K=4) = 128
    v2f a; a.x = Ap[kk * 4]; a.y = Ap[kk * 4 + 1];
    v2f b; b.x = Bp[kk * 4]; b.y = Bp[kk * 4 + 1];
    acc = __builtin_amdgcn_wmma_f32_16x16x4_f32(false, a, false, b,
                                                (short)0, acc, false, false);
  }

  // --- bias + ReLU, spill tile to LDS for the cross-wave LayerNorm
  {
    const float bo  = bproj[ocol];
    const int rbase = (lane < 16) ? 0 : 8;
#pragma unroll
    for (int r = 0; r < 8; ++r) {
      float v = acc[r] + bo;
      v = v > 0.f ? v : 0.f;
      sH[(rbase + r) * 132 + ocol] = v;
    }
  }
  __syncthreads();

  // --- LayerNorm: wave w normalizes rows 2w, 2w+1 (lane-parallel over 128 cols)
#pragma unroll
  for (int rr = 2 * w; rr < 2 * w + 2; ++rr) {
    const float4 v = *(const float4*)(sH + rr * 132 + lane * 4);
    float s1 = v.x + v.y + v.z + v.w;
    float s2 = v.x * v.x + v.y * v.y + v.z * v.z + v.w * v.w;
#pragma unroll
    for (int off = 16; off > 0; off >>= 1) {
      s1 += __shfl_xor(s1, off, 32);
      s2 += __shfl_xor(s2, off, 32);
    }
    const float mu   = s1 * (1.0f / 128.0f);
    const float var  = s2 * (1.0f / 128.0f) - mu * mu;
    const float rstd = rsqrtf(var + LN_EPS);
    const int i = m0 + rr;
    if (i < KEEP) {
      const int c0 = lane * 4;
      const float4 g  = *(const float4*)(gamma + c0);
      const float4 be = *(const float4*)(beta + c0);
      float4 o;
      o.x = (v.x - mu) * rstd * g.x + be.x;
      o.y = (v.y - mu) * rstd * g.y + be.y;
      o.z = (v.z - mu) * rstd * g.z + be.z;
      o.w = (v.w - mu) * rstd * g.w + be.w;
      *(float4*)(xout + (long long)i * DIM + c0) = o;
    }
  }
}

// ---------------------------------------------------------------- edge remap + attr filter
__global__ __launch_bounds__(256) void k8_edges(const int* __restrict__ ei,
                                                const float* __restrict__ ea,
                                                const int* __restrict__ mapping,
                                                int* __restrict__ eio,
                                                float* __restrict__ eao) {
  const int e = blockIdx.x * 256 + threadIdx.x;
  if (e < EDGES) {
    const int s = ei[e];
    const int d = ei[EDGES + e];
    const int r = mapping[s];
    const int c = mapping[d];
    const bool ok = (r >= 0) && (c >= 0);
    eio[e]         = ok ? r : -1;
    eio[EDGES + e] = ok ? c : -1;
    const float4* src = (const float4*)(ea + (long long)e * DEA);
    float4* dst = (float4*)(eao + (long long)e * DEA);
    const float4 z = {0.f, 0.f, 0.f, 0.f};
#pragma unroll
    for (int j = 0; j < 4; ++j) dst[j] = ok ? src[j] : z;
  }
}

// ---------------------------------------------------------------- batch zeros + perm copy
__global__ __launch_bounds__(256) void k9_tail(const int* __restrict__ perm,
                                               int* __restrict__ batcho,
                                               int* __restrict__ permo) {
  const int i = blockIdx.x * 256 + threadIdx.x;
  if (i < KEEP) {
    batcho[i] = 0;
    permo[i]  = perm[i];
  }
}

// ---------------------------------------------------------------- host launcher
extern "C" void kernel_launch(void* const* d_in, const int* in_sizes, int n_in,
                              void* d_out, int out_size, void* d_ws, size_t ws_size,
                              hipStream_t stream) {
  (void)in_sizes; (void)n_in; (void)out_size; (void)ws_size;

  const float* x     = (const float*)d_in[0];
  const int*   ei    = (const int*)d_in[1];
  const float* ea    = (const float*)d_in[2];
  const float* wrel  = (const float*)d_in[3];
  const float* brel  = (const float*)d_in[4];
  const float* wroot = (const float*)d_in[5];
  const float* Wp    = (const float*)d_in[6];
  const float* bproj = (const float*)d_in[7];
  const float* gamma = (const float*)d_in[8];
  const float* beta  = (const float*)d_in[9];

  // workspace layout (256B aligned slabs)
  const size_t SZN = (((size_t)NODES * 4) + 255) & ~(size_t)255;  // 200192
  const size_t SZK = (((size_t)KEEP * 4) + 255) & ~(size_t)255;   // 100096
  char* ws = (char*)d_ws;
  float*    score   = (float*)(ws + 0 * SZN);
  float*    srel    = (float*)(ws + 1 * SZN);
  unsigned* key     = (unsigned*)(ws + 2 * SZN);
  unsigned* ckey    = (unsigned*)(ws + 3 * SZN);
  int*      cidx    = (int*)(ws + 4 * SZN);
  int*      mapping = (int*)(ws + 5 * SZN);
  int*      perm    = (int*)(ws + 6 * SZN);
  float*    tsc     = (float*)(ws + 6 * SZN + SZK);
  unsigned* hist    = (unsigned*)(ws + 6 * SZN + 2 * SZK);
  unsigned* meta    = (unsigned*)(ws + 6 * SZN + 2 * SZK + 16384);

  // output layout (flat, in reference return order)
  float* xout   = (float*)d_out;
  int*   eio    = (int*)d_out + (size_t)KEEP * DIM;
  float* eao    = (float*)d_out + ((size_t)KEEP * DIM + 2 * (size_t)EDGES);
  int*   batcho = (int*)d_out + ((size_t)KEEP * DIM + 2 * (size_t)EDGES + (size_t)EDGES * DEA);
  int*   permo  = batcho + KEEP;

  const dim3 B(256);
  const int GN = (NODES + 255) / 256;   // 196
  const int GE = (EDGES + 255) / 256;   // 2344
  const int G1 = (NODES + 7) / 8;       // 6250 (one wave per node)
  const int G7 = (KEEP + 15) / 16;      // 1563
  const int G9 = (KEEP + 255) / 256;    // 98

  k0_init   <<<GN, B, 0, stream>>>(mapping, hist, meta);
  k1_score  <<<G1, B, 0, stream>>>(x, wrel, wroot, brel, srel, score);
  k2_scatter<<<GE, B, 0, stream>>>(ei, srel, score);
  k3_key    <<<GN, B, 0, stream>>>(score, key, hist);
  k4_select <<<1,  B, 0, stream>>>(hist, meta);
  k5_compact<<<GN, B, 0, stream>>>(key, meta, ckey, cidx);
  k6_rank   <<<GN, B, 0, stream>>>(ckey, cidx, meta, score, perm, tsc, mapping);
  k7_proj   <<<G7, B, 0, stream>>>(x, Wp, bproj, gamma, beta, perm, tsc, xout);
  k8_edges  <<<GE, B, 0, stream>>>(ei, ea, mapping, eio, eao);
  k9_tail   <<<G9, B, 0, stream>>>(perm, batcho, permo);
}